// Model_60533269069826
// MI455X (gfx1250) — compile-verified
//
#include <hip/hip_runtime.h>
#include <stdint.h>

// -------- Kernel 1: per-block dual exclusive scan (span & filter) ----------
#define SCAN_B 256

__global__ void scan_partial(const int* __restrict__ alen,
                             const int* __restrict__ afilt,
                             int nB,
                             int* __restrict__ locOld,   // local excl scan of span
                             int* __restrict__ locNew,   // local excl scan of filter
                             int* __restrict__ sumSpan,  // per-block totals
                             int* __restrict__ sumFilt) {
    __shared__ int s0[SCAN_B];
    __shared__ int s1[SCAN_B];
    const int t = threadIdx.x;
    const int g = blockIdx.x * SCAN_B + t;
    int span = 0, filt = 0;
    if (g < nB) { span = alen[g] + 1; filt = afilt[g]; }
    s0[t] = span; s1[t] = filt;
    __syncthreads();
    // Hillis-Steele inclusive scan on both arrays
    for (int off = 1; off < SCAN_B; off <<= 1) {
        int a0 = (t >= off) ? s0[t - off] : 0;
        int a1 = (t >= off) ? s1[t - off] : 0;
        __syncthreads();
        s0[t] += a0; s1[t] += a1;
        __syncthreads();
    }
    if (g < nB) { locOld[g] = s0[t] - span; locNew[g] = s1[t] - filt; }
    if (t == SCAN_B - 1) { sumSpan[blockIdx.x] = s0[t]; sumFilt[blockIdx.x] = s1[t]; }
}

// -------- Kernel 2: single-block scan of block totals -> exclusive offsets --
__global__ void scan_blocks(int* __restrict__ sumSpan,
                            int* __restrict__ sumFilt,
                            int nblocks) {
    __shared__ int s0[SCAN_B];
    __shared__ int s1[SCAN_B];
    __shared__ int carry0, carry1;
    const int t = threadIdx.x;
    if (t == 0) { carry0 = 0; carry1 = 0; }
    __syncthreads();
    for (int base = 0; base < nblocks; base += SCAN_B) {
        const int g = base + t;
        int v0 = (g < nblocks) ? sumSpan[g] : 0;
        int v1 = (g < nblocks) ? sumFilt[g] : 0;
        s0[t] = v0; s1[t] = v1;
        __syncthreads();
        for (int off = 1; off < SCAN_B; off <<= 1) {
            int a0 = (t >= off) ? s0[t - off] : 0;
            int a1 = (t >= off) ? s1[t - off] : 0;
            __syncthreads();
            s0[t] += a0; s1[t] += a1;
            __syncthreads();
        }
        if (g < nblocks) {
            sumSpan[g] = carry0 + s0[t] - v0;   // exclusive
            sumFilt[g] = carry1 + s1[t] - v1;
        }
        __syncthreads();
        if (t == 0) { carry0 += s0[SCAN_B - 1]; carry1 += s1[SCAN_B - 1]; }
        __syncthreads();
    }
}

// -------- Kernel 3: one wave per batch, async global->LDS->global copy -----
#define WAVES_PER_BLOCK 8
#define MAX_SEG 512   // V = max tokens per request

__global__ void seg_copy_async(const float* __restrict__ src,
                               float* __restrict__ dst,
                               const int* __restrict__ afilt,
                               const int* __restrict__ locOld,
                               const int* __restrict__ locNew,
                               const int* __restrict__ blkOffSpan,
                               const int* __restrict__ blkOffFilt,
                               int nB) {
    __shared__ float buf[WAVES_PER_BLOCK][MAX_SEG];   // 2 KB per wave, 16 KB/block
    const int wave = threadIdx.x >> 5;                // wave32
    const int lane = threadIdx.x & 31;
    const int b = blockIdx.x * WAVES_PER_BLOCK + wave;
    if (b >= nB) return;                              // wave-uniform
    const int len = afilt[b];
    if (len <= 0) return;                             // finished request: skip
    const int os = locOld[b] + blkOffSpan[b >> 8];
    const int ns = locNew[b] + blkOffFilt[b >> 8];

    // wave-relative LDS byte offset of this wave's buffer (flat addr low 32b)
    const unsigned ldsbase = (unsigned)(uintptr_t)(&buf[wave][0]);

    // Phase 1: stream segment HBM -> LDS via async DMA path (ASYNCcnt)
    for (int i = lane; i < len; i += 32) {
        unsigned laddr = ldsbase + (unsigned)i * 4u;
        unsigned long long gaddr = (unsigned long long)(src + os + i);
        asm volatile("global_load_async_to_lds_b32 %0, %1, off"
                     :: "v"(laddr), "v"(gaddr) : "memory");
    }
    asm volatile("s_wait_asynccnt 0" ::: "memory");

    // Phase 2: stream segment LDS -> HBM (compacted destination)
    for (int i = lane; i < len; i += 32) {
        unsigned laddr = ldsbase + (unsigned)i * 4u;
        unsigned long long gaddr = (unsigned long long)(dst + ns + i);
        asm volatile("global_store_async_from_lds_b32 %0, %1, off"
                     :: "v"(gaddr), "v"(laddr) : "memory");
    }
    asm volatile("s_wait_asynccnt 0" ::: "memory");
}

// --------------------------- launcher ---------------------------------------
extern "C" void kernel_launch(void* const* d_in, const int* in_sizes, int n_in,
                              void* d_out, int out_size, void* d_ws, size_t ws_size,
                              hipStream_t stream) {
    const float* tgt_cache_loc = (const float*)d_in[0];
    const int*   accept_length = (const int*)d_in[1];
    const int*   accept_filter = (const int*)d_in[2];
    const int nB = in_sizes[1];                 // 65536
    const int nblocks = (nB + SCAN_B - 1) / SCAN_B;

    int* ws      = (int*)d_ws;
    int* locOld  = ws;                 // [nB]
    int* locNew  = ws + nB;            // [nB]
    int* sumSpan = ws + 2 * nB;        // [nblocks] -> becomes exclusive offsets
    int* sumFilt = ws + 2 * nB + nblocks;

    scan_partial<<<nblocks, SCAN_B, 0, stream>>>(
        accept_length, accept_filter, nB, locOld, locNew, sumSpan, sumFilt);

    scan_blocks<<<1, SCAN_B, 0, stream>>>(sumSpan, sumFilt, nblocks);

    const int copyBlocks = (nB + WAVES_PER_BLOCK - 1) / WAVES_PER_BLOCK;
    seg_copy_async<<<copyBlocks, WAVES_PER_BLOCK * 32, 0, stream>>>(
        tgt_cache_loc, (float*)d_out, accept_filter,
        locOld, locNew, sumSpan, sumFilt, nB);
}